// LPAnalysis_76459007803967
// MI455X (gfx1250) — compile-verified
//
#include <hip/hip_runtime.h>

// LPC residual (ORDER=16, FRAME=160) for MI455X / gfx1250 (wave32).
// One wave == one frame.
//  - Frame staged to LDS via the Tensor Data Mover (tensor_load_to_lds +
//    s_wait_tensorcnt) when available; fallback: global loads + ds stores.
//  - Autocorrelation r[0..15] = row 0 of the 16x16 Gram matrix of the shifted
//    frame, computed with V_WMMA_F32_16X16X4_F32 (full fp32 matrix pipe),
//    using two independent accumulators to halve the XDL dependency depth.
//  - Levinson-Durbin redundantly per lane; 17-tap FIR on VALU.

#define ORDER 16
#define FS 160
#define WAVES_PER_BLOCK 8
#define THREADS (WAVES_PER_BLOCK * 32)
#define FR_STRIDE 192   // 16 front zero-pad + 160 samples + 16 back zero-pad

typedef __attribute__((ext_vector_type(2))) float        v2f;
typedef __attribute__((ext_vector_type(8))) float        v8f;
typedef __attribute__((ext_vector_type(4))) unsigned int v4u;
typedef __attribute__((ext_vector_type(8))) int          v8i;
typedef __attribute__((ext_vector_type(4))) int          v4i;

#if defined(__has_builtin)
#if __has_builtin(__builtin_amdgcn_tensor_load_to_lds) && \
    __has_builtin(__builtin_amdgcn_s_wait_tensorcnt) &&   \
    __has_builtin(__builtin_amdgcn_readfirstlane)
#define USE_TDM 1
#endif
#endif

__global__ __launch_bounds__(THREADS)
void lpc_residual_kernel(const float* __restrict__ x,
                         float* __restrict__ out,
                         int nframes) {
    __shared__ float ldsF[WAVES_PER_BLOCK][FR_STRIDE];
    __shared__ float ldsR[WAVES_PER_BLOCK][20];   // r[0..16] per wave (padded)

    const int lane = threadIdx.x & 31;
    const int w    = threadIdx.x >> 5;
    const int F    = blockIdx.x * WAVES_PER_BLOCK + w;
    const bool active = (F < nframes);

    // Zero the front pad (idx 0..15) and back pad (idx 176..191): one word/lane.
    ldsF[w][(lane < 16) ? lane : (lane + 160)] = 0.0f;

#if USE_TDM
    // ---- TDM: DMA one frame (160 x f32 = 640 B) global -> LDS, per wave ----
    // Descriptor dwords must be wave-uniform SGPRs; TDM ignores EXEC, so gate
    // the issue with a scalar (readfirstlane) branch.
    if (__builtin_amdgcn_readfirstlane((int)active)) {
        unsigned int ldsb =
            (unsigned int)(size_t)&ldsF[w][16];          // low 32 bits = LDS byte offset
        unsigned long long ga =
            (unsigned long long)(size_t)(x + (size_t)F * FS);
        unsigned int ga_lo = (unsigned int)ga;
        unsigned int ga_hi = (unsigned int)(ga >> 32);
        ldsb  = (unsigned int)__builtin_amdgcn_readfirstlane((int)ldsb);
        ga_lo = (unsigned int)__builtin_amdgcn_readfirstlane((int)ga_lo);
        ga_hi = (unsigned int)__builtin_amdgcn_readfirstlane((int)ga_hi);

        // D# group0 (128b): count=1 | lds_addr | global_addr[56:0] | type=2
        v4u g0 = { 1u, ldsb, ga_lo,
                   (ga_hi & 0x01FFFFFFu) | (2u << 30) };
        // D# group1 (256b): data_size=2 (4B); tensor_dim0=160; tensor_dim1=1;
        // tile_dim0=160; tile_dim1=1; tensor_dim0_stride=160; rest 0.
        v8i g1 = { (int)0x00020000u,          // wg_mask=0, data_size=2, flags=0
                   (int)(160u << 16),         // tensor_dim0[15:0] @ bits 63:48
                   (int)(1u << 16),           // dim0[31:16]=0, tensor_dim1[15:0]=1
                   (int)(160u << 16),         // dim1[31:16]=0, tile_dim0=160
                   1,                          // tile_dim1=1, tile_dim2=0
                   160,                        // tensor_dim0_stride[31:0]
                   0, 0 };                     // stride hi, dim1_stride
        v4i z4 = { 0, 0, 0, 0 };               // groups 2/3 unused (2D tensor)
        v8i z8 = { 0, 0, 0, 0, 0, 0, 0, 0 };   // extra group (6-arg toolchain form)
        __builtin_amdgcn_tensor_load_to_lds(g0, g1, z4, z4, z8, 0);
        __builtin_amdgcn_s_wait_tensorcnt(0);
    }
#else
    // ---- Fallback staging: 5 words per lane, global -> VGPR -> LDS ----
    if (active) {
        const float* src = x + (long)F * FS;
#pragma unroll
        for (int k = 0; k < 5; ++k)
            ldsF[w][16 + lane + 32 * k] = src[lane + 32 * k];
    }
#endif
    __syncthreads();

    const float* fB = &ldsF[w][16];   // fB[-16..175] valid; fB[160..175] == 0

    // ---- Gram matrix via WMMA f32 16x16x4: G = X * X^T, X[M][K] = fB[K+M] ----
    // A fragment layout (f32 16x4): lanes 0-15 -> (M=lane, K=v), lanes 16-31 ->
    // (M=lane-16, K=2+v). B (4x16) mirrors it with N across lanes, so for a
    // Gram matrix the A and B register contents are identical.
    v8f acc0, acc1;
#pragma unroll
    for (int i = 0; i < 8; ++i) { acc0[i] = 0.0f; acc1[i] = 0.0f; }

    const int baseLane = (lane < 16) ? lane : (lane - 14);  // lane-16 + 2
#pragma unroll
    for (int c = 0; c < 40; c += 2) {        // K = 160 in chunks of 4, 2-way ILP
        const int b0 = 4 * c + baseLane;
        v2f ab0, ab1;
        ab0.x = fB[b0];     ab0.y = fB[b0 + 1];
        ab1.x = fB[b0 + 4]; ab1.y = fB[b0 + 5];
        acc0 = __builtin_amdgcn_wmma_f32_16x16x4_f32(
            false, ab0, false, ab0, (short)0, acc0, false, false);
        acc1 = __builtin_amdgcn_wmma_f32_16x16x4_f32(
            false, ab1, false, ab1, (short)0, acc1, false, false);
    }

    // r[0..15] = G[0][0..15]: C/D layout VGPR0, lanes 0-15 hold (M=0, N=lane).
    if (lane < 16) ldsR[w][lane] = acc0[0] + acc1[0];

    // r[16] = sum_{t=0}^{143} f[t]*f[t+16]  (branchless mask + wave32 butterfly)
    float p = 0.0f;
#pragma unroll
    for (int k = 0; k < 5; ++k) {
        int t = lane + 32 * k;
        float prod = fB[t] * fB[t + 16];     // fB[175] max -> in bounds
        p += (t < 144) ? prod : 0.0f;
    }
#pragma unroll
    for (int off = 16; off > 0; off >>= 1)
        p += __shfl_xor(p, off, 32);
    if (lane == 0) ldsR[w][16] = p;

    __syncthreads();

    // ---- Levinson-Durbin (redundant per lane; fully unrolled in registers) ----
    float rr[17];
#pragma unroll
    for (int j = 0; j < 17; ++j) rr[j] = ldsR[w][j];

    float a[17];
    a[0] = 1.0f;
#pragma unroll
    for (int j = 1; j < 17; ++j) a[j] = 0.0f;
    float e = rr[0];
#pragma unroll
    for (int i = 1; i <= ORDER; ++i) {
        float s = rr[i];
#pragma unroll
        for (int j = 1; j < i; ++j) s -= a[j] * rr[i - j];
        float kk = s / e;
        float ao[17];
#pragma unroll
        for (int j = 1; j < i; ++j) ao[j] = a[j];
#pragma unroll
        for (int j = 1; j < i; ++j) a[j] = ao[j] - kk * ao[i - j];
        a[i] = kk;
        e *= (1.0f - kk * kk);
    }

    // ---- 17-tap FIR: res[t] = sum_j a[j] * f[t-j] (front pad makes it branchless)
    if (active) {
        float* dst = out + (long)F * FS;
#pragma unroll
        for (int k = 0; k < 5; ++k) {
            int t = lane + 32 * k;
            float s = 0.0f;
#pragma unroll
            for (int j = 0; j <= ORDER; ++j)
                s += a[j] * fB[t - j];       // fB[-16] min -> in bounds (zeros)
            dst[t] = s;
        }
    }
}

extern "C" void kernel_launch(void* const* d_in, const int* in_sizes, int n_in,
                              void* d_out, int out_size, void* d_ws, size_t ws_size,
                              hipStream_t stream) {
    const float* x = (const float*)d_in[0];
    float* out = (float*)d_out;
    const int total = in_sizes[0];           // 32 * 160000 = 5,120,000
    const int nframes = total / FS;          // 32,000
    const int blocks = (nframes + WAVES_PER_BLOCK - 1) / WAVES_PER_BLOCK;  // 4000
    lpc_residual_kernel<<<blocks, THREADS, 0, stream>>>(x, out, nframes);
}